// CGAMixer_11579231830738
// MI455X (gfx1250) — compile-verified
//
#include <hip/hip_runtime.h>
#include <hip/hip_bf16.h>
#include <math.h>

// ---------------- WMMA types (CDNA5 / gfx1250, wave32) ----------------
typedef __attribute__((ext_vector_type(2))) float v2f;
typedef __attribute__((ext_vector_type(8))) float v8f;

#define D_MODEL 256
#define D_STATE 64
#define MAXC 64
#define NEG_INF (-1e9f)

// =====================================================================
// FP32 WMMA GEMM:  C[M x N] = A[M x K] @ W[K x N] + bias
// One wave computes one 16x16 tile via V_WMMA_F32_16X16X4_F32, K-loop by 4.
// A-matrix lane layout (ISA 7.12.2, 32-bit A 16x4):
//   lane l: m = l&15 ; VGPR0 -> k = 2*(l>>4), VGPR1 -> k = 2*(l>>4)+1
// B-matrix (4x16): lane l: n = l&15 ; VGPR0 -> k = 2*(l>>4), VGPR1 -> +1
// D (16x16 f32): vgpr i, lane l -> (m = i + 8*(l>>4), n = l&15)
// =====================================================================
__global__ void __launch_bounds__(128)
wmma_gemm_f32(const float* __restrict__ A, const float* __restrict__ W,
              const float* __restrict__ bias, float* __restrict__ C,
              int M, int N, int K) {
  const int wavesPerBlock = blockDim.x >> 5;
  const int tile = blockIdx.x * wavesPerBlock + (threadIdx.x >> 5);
  const int lane = threadIdx.x & 31;
  const int tilesN = N >> 4;
  const int tm = tile / tilesN;
  const int tn = tile - tm * tilesN;
  if (tm >= (M >> 4)) return;          // wave-uniform exit (EXEC stays all-1s)

  const int m0 = tm << 4;
  const int n0 = tn << 4;
  const int mrow = m0 + (lane & 15);
  const int ncol = n0 + (lane & 15);
  const int kb = (lane >> 4) << 1;     // 0 or 2

  v8f acc = {};
  for (int k0 = 0; k0 < K; k0 += 4) {
    v2f a, b;
    const float* ap = A + (size_t)mrow * K + (k0 + kb);
    a.x = ap[0];
    a.y = ap[1];
    const float* bp = W + (size_t)(k0 + kb) * N + ncol;
    b.x = bp[0];
    b.y = bp[N];
    // 8 args: (neg_a, A, neg_b, B, c_mod, C, reuse_a, reuse_b)
    acc = __builtin_amdgcn_wmma_f32_16x16x4_f32(
        false, a, false, b, (short)0, acc, false, false);
  }

  const float bn = bias ? bias[ncol] : 0.0f;
  const int mbase = m0 + ((lane >> 4) << 3);
  #pragma unroll
  for (int i = 0; i < 8; ++i) {
    C[(size_t)(mbase + i) * N + ncol] = acc[i] + bn;
  }
}

// =====================================================================
// Row-wise L2 normalize for 64-wide rows (q/k). One wave per row (wave32:
// each lane owns 2 elements). eps = 1e-12 as in F.normalize.
// =====================================================================
__global__ void __launch_bounds__(256)
l2norm_rows64(float* __restrict__ data, int rows) {
  const int wid = (blockIdx.x * blockDim.x + threadIdx.x) >> 5;
  const int lane = threadIdx.x & 31;
  if (wid >= rows) return;
  float* row = data + (size_t)wid * 64;
  float a = row[lane];
  float b = row[lane + 32];
  float ss = a * a + b * b;
  #pragma unroll
  for (int off = 16; off > 0; off >>= 1) ss += __shfl_xor(ss, off, 32);
  const float inv = 1.0f / fmaxf(sqrtf(ss), 1e-12f);
  row[lane] = a * inv;
  row[lane + 32] = b * inv;
}

// =====================================================================
// Sequential concept-memory scan. One block per batch sequence; the entire
// concept state (centroids 64x64, values 64x256, counts) lives in LDS
// (~84 KB -> needs CDNA5's 320KB-per-WGP LDS; allocated dynamically).
// 256 threads: lanes 0..63 own centroid columns, all 256 own value columns.
// =====================================================================
__global__ void __launch_bounds__(256)
concept_scan(const float* __restrict__ q, const float* __restrict__ k,
             const float* __restrict__ v, float* __restrict__ y,
             const float* __restrict__ logit_scale_p, int T) {
  extern __shared__ float smem[];
  float* s_cent = smem;                    // 64*64
  float* s_vals = s_cent + MAXC * D_STATE; // 64*256
  float* s_cnt  = s_vals + MAXC * D_MODEL; // 64
  float* s_q    = s_cnt + MAXC;            // 64
  float* s_k    = s_q + D_STATE;           // 64
  float* s_v    = s_k + D_STATE;           // 256
  float* s_sim  = s_v + D_MODEL;           // 64
  float* s_w    = s_sim + MAXC;            // 64 (scores -> exp weights)
  float* s_red  = s_w + MAXC;              // 256 reduction scratch

  __shared__ int   s_n, s_sel, s_doAdd, s_doUpd;
  __shared__ float s_maxv, s_invsum, s_cnorm;

  const int tid = threadIdx.x;
  const float scale = fminf(__expf(logit_scale_p[0]), 100.0f);

  for (int i = tid; i < MAXC * D_STATE; i += 256) s_cent[i] = 0.0f;
  for (int i = tid; i < MAXC * D_MODEL; i += 256) s_vals[i] = 0.0f;
  if (tid < MAXC) s_cnt[tid] = 0.0f;
  if (tid == 0) s_n = 0;
  __syncthreads();

  for (int t = 0; t < T; ++t) {
    const size_t row = (size_t)blockIdx.x * T + t;
    if (tid < D_STATE) {
      s_q[tid] = q[row * D_STATE + tid];
      s_k[tid] = k[row * D_STATE + tid];
    }
    s_v[tid] = v[row * D_MODEL + tid];
    __syncthreads();

    const int n = s_n;

    // ---- sims = centroids @ q_t ; masked scores ----
    if (tid < MAXC) {
      float s = 0.0f;
      const float* c = s_cent + tid * D_STATE;
      #pragma unroll 8
      for (int j = 0; j < D_STATE; ++j) s = fmaf(c[j], s_q[j], s);
      s_sim[tid] = s;
      s_w[tid] = (tid < n) ? s * scale : NEG_INF * scale;
    }
    __syncthreads();

    // ---- max + argmax (first-index ties, matching jnp.argmax) ----
    if (tid < 32) {
      float v0 = s_w[tid], v1 = s_w[tid + 32];
      float bv = (v1 > v0) ? v1 : v0;
      int bi = (v1 > v0) ? (tid + 32) : tid;
      #pragma unroll
      for (int off = 16; off > 0; off >>= 1) {
        float ov = __shfl_xor(bv, off, 32);
        int oi = __shfl_xor(bi, off, 32);
        if (ov > bv || (ov == bv && oi < bi)) { bv = ov; bi = oi; }
      }
      if (tid == 0) { s_maxv = bv; s_sel = bi; }
    }
    __syncthreads();

    // ---- softmax weights ----
    if (tid < MAXC) s_w[tid] = __expf(s_w[tid] - s_maxv);
    __syncthreads();
    if (tid < 32) {
      float sum = s_w[tid] + s_w[tid + 32];
      #pragma unroll
      for (int off = 16; off > 0; off >>= 1) sum += __shfl_xor(sum, off, 32);
      if (tid == 0) s_invsum = 1.0f / sum;
    }
    __syncthreads();

    const int sel = s_sel;

    // ---- z = softmax(w) @ values (uses pre-update values) ----
    {
      float zd = 0.0f;
      #pragma unroll 8
      for (int m = 0; m < MAXC; ++m)
        zd = fmaf(s_w[m], s_vals[m * D_MODEL + tid], zd);
      zd *= s_invsum;
      y[row * D_MODEL + tid] = (n > 0) ? zd : 0.0f;
    }

    // ---- residual = sqrt(mean((values[sel] - v_t)^2)) ----
    {
      const float d = s_vals[sel * D_MODEL + tid] - s_v[tid];
      s_red[tid] = d * d;
    }
    __syncthreads();
    if (tid < 128) s_red[tid] += s_red[tid + 128];
    __syncthreads();
    if (tid < 64) s_red[tid] += s_red[tid + 64];
    __syncthreads();
    if (tid < 32) {
      float r = s_red[tid] + s_red[tid + 32];
      #pragma unroll
      for (int off = 16; off > 0; off >>= 1) r += __shfl_xor(r, off, 32);
      if (tid == 0) {
        const float residual = sqrtf(r * (1.0f / D_MODEL));
        const float sel_sim = s_sim[sel];
        const bool has = (n > 0);
        const bool refine = has && (n < MAXC) &&
                            ((sel_sim < 0.75f) || (residual > 1.0f));
        s_doAdd = (((!has) || refine) && (n < MAXC)) ? 1 : 0;
        s_doUpd = (has && !refine) ? 1 : 0;
      }
    }
    __syncthreads();
    const int doAdd = s_doAdd;
    const int doUpd = s_doUpd;

    // ---- refine existing concept (block-uniform branch) ----
    if (doUpd) {
      const float cnt = s_cnt[sel];                       // read before write
      const float vold = s_vals[sel * D_MODEL + tid];
      float nc = 0.0f;
      if (tid < D_STATE) {
        nc = 0.9f * s_cent[sel * D_STATE + tid] + 0.1f * s_k[tid];
        s_red[tid] = nc * nc;
      }
      __syncthreads();
      if (tid < 32) {
        float ss = s_red[tid] + s_red[tid + 32];
        #pragma unroll
        for (int off = 16; off > 0; off >>= 1) ss += __shfl_xor(ss, off, 32);
        if (tid == 0) s_cnorm = 1.0f / fmaxf(sqrtf(ss), 1e-12f);
      }
      __syncthreads();
      if (tid < D_STATE) s_cent[sel * D_STATE + tid] = nc * s_cnorm;
      s_vals[sel * D_MODEL + tid] = (vold * cnt + s_v[tid]) / (cnt + 1.0f);
      if (tid == 0) s_cnt[sel] = cnt + 1.0f;
    }

    // ---- create new concept in slot n (sel < n, no collision) ----
    if (doAdd) {
      const int slot = (n < MAXC - 1) ? n : (MAXC - 1);
      if (tid < D_STATE) s_cent[slot * D_STATE + tid] = s_k[tid];
      s_vals[slot * D_MODEL + tid] = s_v[tid];
      if (tid == 0) { s_cnt[slot] = 1.0f; s_n = n + 1; }
    }
    __syncthreads();
  }
}

// =====================================================================
// Host-side launcher
// =====================================================================
extern "C" void kernel_launch(void* const* d_in, const int* in_sizes, int n_in,
                              void* d_out, int out_size, void* d_ws, size_t ws_size,
                              hipStream_t stream) {
  (void)n_in; (void)out_size; (void)ws_size;
  const float* x  = (const float*)d_in[0];
  const float* Wq = (const float*)d_in[1];
  const float* bq = (const float*)d_in[2];
  const float* Wk = (const float*)d_in[3];
  const float* bk = (const float*)d_in[4];
  const float* Wv = (const float*)d_in[5];
  const float* bv = (const float*)d_in[6];
  const float* Wo = (const float*)d_in[7];
  const float* bo = (const float*)d_in[8];
  const float* logit_scale = (const float*)d_in[9];
  float* out = (float*)d_out;

  const int B = 4, T = 1024;
  const int BT = in_sizes[0] / D_MODEL;   // 4096

  // Workspace layout (fp32): q | k | v | y
  float* q = (float*)d_ws;
  float* k = q + (size_t)BT * D_STATE;
  float* v = k + (size_t)BT * D_STATE;
  float* y = v + (size_t)BT * D_MODEL;

  const int block = 128;                  // 4 waves/block, 1 tile per wave
  const int wavesPerBlock = block / 32;

  // q = x @ Wq + bq   (M=BT, N=64, K=256)
  {
    int tiles = (BT / 16) * (D_STATE / 16);
    wmma_gemm_f32<<<(tiles + wavesPerBlock - 1) / wavesPerBlock, block, 0, stream>>>(
        x, Wq, bq, q, BT, D_STATE, D_MODEL);
  }
  // k = x @ Wk + bk
  {
    int tiles = (BT / 16) * (D_STATE / 16);
    wmma_gemm_f32<<<(tiles + wavesPerBlock - 1) / wavesPerBlock, block, 0, stream>>>(
        x, Wk, bk, k, BT, D_STATE, D_MODEL);
  }
  // v = x @ Wv + bv   (M=BT, N=256, K=256)
  {
    int tiles = (BT / 16) * (D_MODEL / 16);
    wmma_gemm_f32<<<(tiles + wavesPerBlock - 1) / wavesPerBlock, block, 0, stream>>>(
        x, Wv, bv, v, BT, D_MODEL, D_MODEL);
  }

  // l2-normalize q and k rows (eps 1e-12)
  {
    int waves = BT;                       // one wave per row
    int blocks = (waves * 32 + 255) / 256;
    l2norm_rows64<<<blocks, 256, 0, stream>>>(q, BT);
    l2norm_rows64<<<blocks, 256, 0, stream>>>(k, BT);
  }

  // sequential concept scan: one block per sequence, big dynamic LDS
  {
    const size_t smemFloats =
        (size_t)MAXC * D_STATE +   // centroids
        (size_t)MAXC * D_MODEL +   // values
        MAXC +                     // counts
        D_STATE + D_STATE +        // q_t, k_t
        D_MODEL +                  // v_t
        MAXC + MAXC +              // sims, weights
        D_MODEL;                   // reduction scratch
    concept_scan<<<B, 256, smemFloats * sizeof(float), stream>>>(
        q, k, v, y, logit_scale, T);
  }

  // out = y @ Wo + bo
  {
    int tiles = (BT / 16) * (D_MODEL / 16);
    wmma_gemm_f32<<<(tiles + wavesPerBlock - 1) / wavesPerBlock, block, 0, stream>>>(
        y, Wo, bo, out, BT, D_MODEL, D_MODEL);
  }
}